// HungarianMatcher_13941463843270
// MI455X (gfx1250) — compile-verified
//
#include <hip/hip_runtime.h>
#include <math.h>

// DETR HungarianMatcher cost matrix on gfx1250 (MI455X).
// cost_class as one-hot GEMM via V_WMMA_F32_16X16X4_F32 (bit-exact gather).
// One block = 16 rows x 512 cols; one wave = 16 x 64 (4 WMMA accumulators).

typedef __attribute__((ext_vector_type(2))) float v2f;
typedef __attribute__((ext_vector_type(4))) float v4f;
typedef __attribute__((ext_vector_type(8))) float v8f;

static constexpr int NC = 92;               // classes (K dim), 92 % 4 == 0
static constexpr int TW = 8;                // waves per block
static constexpr int CW = 64;               // cols per wave (4 WMMA tiles)
static constexpr int CB = TW * CW;          // cols per block = 512

__device__ __forceinline__ float fastdiv(float a, float b) {
  return a * __builtin_amdgcn_rcpf(b);      // v_rcp_f32 + v_mul (~1 ulp rcp)
}

__global__ __launch_bounds__(256)
void detr_cost_kernel(const float* __restrict__ logits,   // [nrows, NC]
                      const float* __restrict__ pboxes,   // [nrows, 4] cxcywh
                      const float* __restrict__ tboxes,   // [M, 4] cxcywh
                      const int*   __restrict__ tlabels,  // [M]
                      float* __restrict__ out,            // [nrows, M]
                      int nrows, int M) {
  __shared__ float ptile[16 * NC];    // 5888 B: this block's 16-row prob tile

  const int t    = threadIdx.x;
  const int lane = t & 31;
  const int wv   = t >> 5;
  const int rowbase = blockIdx.y << 4;
  if (rowbase + 16 > nrows) return;   // block-uniform

  // ---- stage 16 x NC logits into LDS (coalesced) ----
  const float* gsrc = logits + (size_t)rowbase * NC;
  for (int i = t; i < 16 * NC; i += 256) ptile[i] = gsrc[i];
  __syncthreads();

  // ---- softmax: wave wv owns rows 2wv, 2wv+1 (reduce + convert in place).
  //      After the shfl butterfly every lane holds the full max/sum, so the
  //      owning wave converts its rows from registers: no stats LDS, 1 barrier.
#pragma unroll
  for (int rr = 0; rr < 2; ++rr) {
    float* x = &ptile[((wv << 1) + rr) * NC];
    float m = -3.402823466e+38f;
    for (int c = lane; c < NC; c += 32) m = fmaxf(m, x[c]);
    for (int off = 16; off; off >>= 1) m = fmaxf(m, __shfl_xor(m, off, 32));
    float s = 0.0f;
    for (int c = lane; c < NC; c += 32) s += __expf(x[c] - m);
    for (int off = 16; off; off >>= 1) s += __shfl_xor(s, off, 32);
    const float inv = fastdiv(1.0f, s);
    for (int c = lane; c < NC; c += 32) x[c] = __expf(x[c] - m) * inv;
  }
  __syncthreads();

  // ---- per-wave 16 x 64 output tile ----
  const int lh = lane & 15;           // A row / B col / D col within 16-tile
  const int hi = lane >> 4;           // half-wave selector
  const int colbase = blockIdx.x * CB + wv * CW;
  if (colbase >= M) return;           // wave-uniform, after all barriers

  // Clipped labels, pre-shifted by the half-wave K offset so the inner loop
  // compares against literal k0 / k0+1 (no per-iter ka construction).
  int tda[4];
#pragma unroll
  for (int g = 0; g < 4; ++g) {
    int c = tlabels[colbase + 16 * g + lh];
    c = c < 0 ? 0 : (c > NC - 1 ? NC - 1 : c);
    tda[g] = c - (hi << 1);
  }

  // K-loop: one A fragment feeds 4 WMMAs (one per column group).
  // A layout (16x4 f32): lane L holds row L%16; vgpr j -> K = j + 2*(L>=16).
  // B layout (mirrored): lane L holds col L%16; vgpr j -> K = j + 2*(L>=16).
  v8f acc[4] = {{}, {}, {}, {}};
  const int rowoff = lh * NC;
#pragma unroll
  for (int k0 = 0; k0 < NC; k0 += 4) {
    v2f a;
    a.x = ptile[rowoff + k0 + (hi << 1)];
    a.y = ptile[rowoff + k0 + (hi << 1) + 1];
#pragma unroll
    for (int g = 0; g < 4; ++g) {
      v2f b;
      b.x = (tda[g] == k0)     ? 1.0f : 0.0f;   // literal-constant compares
      b.y = (tda[g] == k0 + 1) ? 1.0f : 0.0f;
      acc[g] = __builtin_amdgcn_wmma_f32_16x16x4_f32(
          false, a, false, b, (short)0, acc[g], false, false);
    }
  }

  // ---- target-box precompute per column group ----
  v4f   tb[4];
  float tx0[4], ty0[4], tx1[4], ty1[4], areab[4];
#pragma unroll
  for (int g = 0; g < 4; ++g) {
    tb[g] = *(const v4f*)(tboxes + (size_t)(colbase + 16 * g + lh) * 4);
    tx0[g] = tb[g].x - 0.5f * tb[g].z;  ty0[g] = tb[g].y - 0.5f * tb[g].w;
    tx1[g] = tb[g].x + 0.5f * tb[g].z;  ty1[g] = tb[g].y + 0.5f * tb[g].w;
    areab[g] = (tx1[g] - tx0[g]) * (ty1[g] - ty0[g]);
  }

  // ---- epilogue: D element acc[g][r] is (row = rowbase + r + 8*hi,
  //                                        col = colbase + 16*g + lh) ----
#pragma unroll
  for (int r = 0; r < 8; ++r) {
    const int row = rowbase + r + (hi << 3);
    const v4f pb = *(const v4f*)(pboxes + (size_t)row * 4);
    const float ax0 = pb.x - 0.5f * pb.z, ay0 = pb.y - 0.5f * pb.w;
    const float ax1 = pb.x + 0.5f * pb.z, ay1 = pb.y + 0.5f * pb.w;
    const float area_a = (ax1 - ax0) * (ay1 - ay0);
    float* orow = out + (size_t)row * M + colbase + lh;
#pragma unroll
    for (int g = 0; g < 4; ++g) {
      // L1 cdist on cxcywh.
      const float cbbox = fabsf(pb.x - tb[g].x) + fabsf(pb.y - tb[g].y) +
                          fabsf(pb.z - tb[g].z) + fabsf(pb.w - tb[g].w);
      // GIoU on xyxy.
      const float ix0 = fmaxf(ax0, tx0[g]), iy0 = fmaxf(ay0, ty0[g]);
      const float ix1 = fminf(ax1, tx1[g]), iy1 = fminf(ay1, ty1[g]);
      const float iw = fmaxf(ix1 - ix0, 0.0f), ih = fmaxf(iy1 - iy0, 0.0f);
      const float inter = iw * ih;
      const float uni = area_a + areab[g] - inter;
      const float iou = fastdiv(inter, uni);
      // Enclosing box: ex1 >= ex0 and ey1 >= ey0 always (max >= min), so the
      // reference's clip-at-0 is an identity and is elided.
      const float ex0 = fminf(ax0, tx0[g]), ey0 = fminf(ay0, ty0[g]);
      const float ex1 = fmaxf(ax1, tx1[g]), ey1 = fmaxf(ay1, ty1[g]);
      const float area_e = (ex1 - ex0) * (ey1 - ey0);
      const float giou = iou - fastdiv(area_e - uni, area_e);
      // C = 5*bbox + 1*(-prob) + 2*(-giou)
      orow[16 * g] = 5.0f * cbbox - acc[g][r] - 2.0f * giou;
    }
  }
}

// ---------------------------------------------------------------------------
extern "C" void kernel_launch(void* const* d_in, const int* in_sizes, int n_in,
                              void* d_out, int out_size, void* d_ws, size_t ws_size,
                              hipStream_t stream) {
  const float* logits  = (const float*)d_in[0];  // [BS,NQ,NC] f32
  const float* pboxes  = (const float*)d_in[1];  // [BS,NQ,4]  f32
  const float* tboxes  = (const float*)d_in[2];  // [M,4]      f32
  const int*   tlabels = (const int*)d_in[3];    // [M]        i32

  const int nrows = in_sizes[1] / 4;  // BS*NQ = 16000
  const int M     = in_sizes[3];      // 1024
  float* out = (float*)d_out;

  dim3 block(256);
  dim3 grid((M + CB - 1) / CB, (nrows + 15) / 16);   // (2, 1000)
  detr_cost_kernel<<<grid, block, 0, stream>>>(logits, pboxes, tboxes, tlabels,
                                               out, nrows, M);
}